// LiteFlowNetCorrelation_27771258536138
// MI455X (gfx1250) — compile-verified
//
#include <hip/hip_runtime.h>
#include <hip/hip_bf16.h>

// CDNA5 / gfx1250 correlation kernel using V_WMMA_F32_16X16X4_F32 (fp32 WMMA).
//
// Gram-band formulation: for fixed (b, h, row-shift i), the band
//   out[w, j] = sum_c x1[c,w] * x2_row[c, w+j-4],  j = 0..8
// is a diagonal band of the Gram matrix G = X1tile^T * X2tile, which we
// compute as two 16x16 WMMA output tiles (N spans y = w0-4 .. w0+27).

typedef __attribute__((ext_vector_type(2))) float v2f;
typedef __attribute__((ext_vector_type(8))) float v8f;

#define B_   8
#define C_   128
#define H_   96
#define W_   160
#define HW_  (H_ * W_)
#define ND_  9            // 2d+1
#define WT_  (W_ / 16)    // 10 w-tiles
#define WPB_ 8            // waves per 256-thread block

__global__ __launch_bounds__(256)
void corr_wmma_kernel(const float* __restrict__ x1,
                      const float* __restrict__ x2,
                      float* __restrict__ out)
{
    // Per-wave 16x32 Gram staging tile, padded to 34 cols for bank-conflict-free
    // row writes and diagonal reads.
    __shared__ float lds[WPB_][16][34];

    const int lane  = threadIdx.x & 31;
    const int wv    = threadIdx.x >> 5;
    const int m     = lane & 15;     // pixel within tile / N within tile
    const int chalf = lane >> 4;     // K-half select (ISA 16x4 f32 A layout)

    int gw = blockIdx.x * WPB_ + wv;
    const int i  = gw % ND_; gw /= ND_;   // row displacement 0..8
    const int wt = gw % WT_; gw /= WT_;
    const int h  = gw % H_;  gw /= H_;
    const int b  = gw;                    // 0..7

    const int w0 = wt * 16;
    const int h2 = h + i - 4;             // x2 source row (may be padded)

    float* __restrict__ outp = out + ((b * 81 + i * ND_) * H_ + h) * W_ + w0;

    if (h2 < 0 || h2 >= H_) {
        // Entire x2 row is zero padding -> all 9 j-planes are zero here.
        for (int j = chalf; j < ND_; j += 2)
            outp[j * HW_ + m] = 0.0f;
        return;
    }

    const int in_b = b * C_ * HW_;

    // A fragment base: A[m, k] = x1[c, w0+m]; lane m = pixel, channel offsets
    // become immediate offsets after full unroll (max 127*HW_*4B = 7.8MB < 8MB).
    const float* __restrict__ pA = x1 + in_b + h * W_ + (w0 + m);

    // B fragments: tile0 covers y = w0-4+n, tile1 covers y = w0+12+n (n = m).
    const int  y0 = w0 - 4 + m;
    const int  y1 = y0 + 16;
    const bool v0 = (unsigned)y0 < (unsigned)W_;   // left-edge padding
    const bool v1 = (unsigned)y1 < (unsigned)W_;   // right-edge padding
    const float* __restrict__ pB0 = x2 + in_b + h2 * W_ + (v0 ? y0 : 0);
    const float* __restrict__ pB1 = x2 + in_b + h2 * W_ + (v1 ? y1 : 0);

    v8f acc0 = {};   // G[:, 0..15]
    v8f acc1 = {};   // G[:, 16..31]

#pragma unroll
    for (int kk = 0; kk < C_ / 4; ++kk) {
        const int c0 = kk * 4 + 2 * chalf;   // this lane's two K channels
        v2f a, bb0, bb1;
        a.x = pA[c0 * HW_];
        a.y = pA[(c0 + 1) * HW_];
        // Unconditional loads from clamped (always valid) pointers, then
        // branch-free zero-select for padded columns (v_cndmask, EXEC stays all-1s).
        const float t00 = pB0[c0 * HW_];
        const float t01 = pB0[(c0 + 1) * HW_];
        const float t10 = pB1[c0 * HW_];
        const float t11 = pB1[(c0 + 1) * HW_];
        bb0.x = v0 ? t00 : 0.0f;  bb0.y = v0 ? t01 : 0.0f;
        bb1.x = v1 ? t10 : 0.0f;  bb1.y = v1 ? t11 : 0.0f;

        // D = A*B + C ; 8 args: (neg_a, A, neg_b, B, c_mod, C, reuse_a, reuse_b)
        acc0 = __builtin_amdgcn_wmma_f32_16x16x4_f32(
            false, a, false, bb0, (short)0, acc0, false, false);
        acc1 = __builtin_amdgcn_wmma_f32_16x16x4_f32(
            false, a, false, bb1, (short)0, acc1, false, false);
    }

    // Stage G[m][n] (n = 0..31) into LDS. D layout: VGPR v holds row
    // m = v + 8*(lane>=16), column n = lane&15.
#pragma unroll
    for (int v = 0; v < 8; ++v) {
        const int row = v + 8 * chalf;
        lds[wv][row][m]      = acc0[v];
        lds[wv][row][16 + m] = acc1[v];
    }
    // Same-wave DS store -> DS load ordering.
    asm volatile("s_wait_dscnt 0" ::: "memory");

    // Extract diagonal band: out[w0+m, j] = G[m][m+j] / C. Lanes 0-15 handle
    // even j, lanes 16-31 odd j -> 16-wide coalesced stores per j-plane.
    const float scale = 1.0f / (float)C_;
#pragma unroll
    for (int t = 0; t < 5; ++t) {
        const int j = 2 * t + chalf;
        if (j < ND_) {
            outp[j * HW_ + m] = lds[wv][m][m + j] * scale;
        }
    }
}

extern "C" void kernel_launch(void* const* d_in, const int* in_sizes, int n_in,
                              void* d_out, int out_size, void* d_ws, size_t ws_size,
                              hipStream_t stream)
{
    const float* x1 = (const float*)d_in[0];
    const float* x2 = (const float*)d_in[1];
    // d_in[2] is the displacement radius d (==4), baked into the kernel.
    float* out = (float*)d_out;

    // waves = B * H * WT * ND = 8*96*10*9 = 69120 ; 8 waves per block.
    const int total_waves = B_ * H_ * WT_ * ND_;
    const int blocks = total_waves / WPB_;   // 8640
    hipLaunchKernelGGL(corr_wmma_kernel, dim3(blocks), dim3(256), 0, stream,
                       x1, x2, out);
}